// Loss_yolov1_63909113365285
// MI455X (gfx1250) — compile-verified
//
#include <hip/hip_runtime.h>

typedef float v2f __attribute__((ext_vector_type(2)));
typedef float v4f __attribute__((ext_vector_type(4)));
typedef float v8f __attribute__((ext_vector_type(8)));

#define NBLOCKS 1024
#define BLOCK   256
#define NWAVES  (BLOCK / 32)

// ---------------------------------------------------------------------------
// Per-cell YOLOv1 loss terms. p/t point at the 30-float records for one cell.
// Components: loc, contain, nooobj, class, not_contain.
// ---------------------------------------------------------------------------
__device__ __forceinline__ void cell_loss(const float* __restrict__ p,
                                          const float* __restrict__ t,
                                          float& loc, float& con, float& noo_s,
                                          float& cls, float& ncf) {
  const float inv = 1.0f / 14.0f;

  float conf_t = t[4];
  float coo = (conf_t > 0.f) ? 1.f : 0.f;
  float noo = (conf_t == 0.f) ? 1.f : 0.f;

  float d4 = p[4] - t[4], d9 = p[9] - t[9];
  noo_s += noo * (d4 * d4 + d9 * d9);

  // target box 0 in xyxy
  float tcx = t[0] * inv, tcy = t[1] * inv;
  float tx0 = tcx - 0.5f * t[2], ty0 = tcy - 0.5f * t[3];
  float tx1 = tcx + 0.5f * t[2], ty1 = tcy + 0.5f * t[3];
  float ta = (tx1 - tx0) * (ty1 - ty0);

  float iou[2];
#pragma unroll
  for (int k = 0; k < 2; ++k) {
    const float* b = p + 5 * k;
    float cx = b[0] * inv, cy = b[1] * inv;
    float x0 = cx - 0.5f * b[2], y0 = cy - 0.5f * b[3];
    float x1 = cx + 0.5f * b[2], y1 = cy + 0.5f * b[3];
    float lx = fmaxf(x0, tx0), ly = fmaxf(y0, ty0);
    float rx = fminf(x1, tx1), ry = fminf(y1, ty1);
    float w = fmaxf(rx - lx, 0.f), h = fmaxf(ry - ly, 0.f);
    float inter = w * h;
    float pa = (x1 - x0) * (y1 - y0);
    iou[k] = inter / (pa + ta - inter);
  }

  // jnp.argmax: first index wins on tie -> idx=1 only if strictly greater
  int idx = (iou[1] > iou[0]) ? 1 : 0;
  float miou = idx ? iou[1] : iou[0];
  const float* rp = p + 5 * idx;
  const float* rt = t + 5 * idx;
  float nrc = idx ? p[4] : p[9];  // conf of the non-responsible pred box

  float dc = rp[4] - miou;
  con += coo * dc * dc;

  float dx = rp[0] - rt[0], dy = rp[1] - rt[1];
  float dw = sqrtf(rp[2]) - sqrtf(rt[2]);
  float dh = sqrtf(rp[3]) - sqrtf(rt[3]);
  loc += coo * (dx * dx + dy * dy + dw * dw + dh * dh);

  ncf += coo * nrc * nrc;

  float cs = 0.f;
#pragma unroll
  for (int c2 = 10; c2 < 30; ++c2) {
    float d = p[c2] - t[c2];
    cs += d * d;
  }
  cls += coo * cs;
}

// ---------------------------------------------------------------------------
// Kernel 1: streaming loss, two cells per thread-iteration so each 240 B
// record-pair is 16 B aligned -> non-temporal global_load_b128 (TH=NT; data
// is read exactly once and the 192 MB footprint equals all of L2).
// Hierarchical deterministic reduce -> ws[c*NBLOCKS + blockIdx.x].
// ---------------------------------------------------------------------------
__global__ void yolo_partial(const float* __restrict__ pred,
                             const float* __restrict__ targ,
                             float* __restrict__ ws, int ncells) {
  float loc = 0.f, con = 0.f, noo_s = 0.f, cls = 0.f, ncf = 0.f;

  int npairs = ncells >> 1;
  int gid = blockIdx.x * blockDim.x + threadIdx.x;
  int stride = gridDim.x * blockDim.x;

  for (int pair = gid; pair < npairs; pair += stride) {
    const v4f* pp = reinterpret_cast<const v4f*>(pred) + (size_t)pair * 15;
    const v4f* tp = reinterpret_cast<const v4f*>(targ) + (size_t)pair * 15;
    float p[60], t[60];
#pragma unroll
    for (int k = 0; k < 15; ++k) {
      v4f a = __builtin_nontemporal_load(pp + k);
      p[4 * k + 0] = a.x; p[4 * k + 1] = a.y;
      p[4 * k + 2] = a.z; p[4 * k + 3] = a.w;
      v4f b = __builtin_nontemporal_load(tp + k);
      t[4 * k + 0] = b.x; t[4 * k + 1] = b.y;
      t[4 * k + 2] = b.z; t[4 * k + 3] = b.w;
    }
#pragma unroll
    for (int h = 0; h < 2; ++h)
      cell_loss(p + 30 * h, t + 30 * h, loc, con, noo_s, cls, ncf);
  }

  // odd-cell tail (not hit for B*49 even, kept for generality)
  if ((ncells & 1) && gid == 0) {
    const float2* pp2 =
        reinterpret_cast<const float2*>(pred) + (size_t)(ncells - 1) * 15;
    const float2* tp2 =
        reinterpret_cast<const float2*>(targ) + (size_t)(ncells - 1) * 15;
    float p[30], t[30];
#pragma unroll
    for (int k = 0; k < 15; ++k) {
      float2 a = pp2[k]; p[2 * k] = a.x; p[2 * k + 1] = a.y;
      float2 b = tp2[k]; t[2 * k] = b.x; t[2 * k + 1] = b.y;
    }
    cell_loss(p, t, loc, con, noo_s, cls, ncf);
  }

  // wave32 shuffle-tree reduction (deterministic)
  float v[5] = {loc, con, noo_s, cls, ncf};
#pragma unroll
  for (int c = 0; c < 5; ++c)
#pragma unroll
    for (int off = 16; off > 0; off >>= 1)
      v[c] += __shfl_down(v[c], off, 32);

  __shared__ float sm[NWAVES][5];
  int lane = threadIdx.x & 31;
  int wv = threadIdx.x >> 5;
  if (lane == 0) {
#pragma unroll
    for (int c = 0; c < 5; ++c) sm[wv][c] = v[c];
  }
  __syncthreads();
  if (threadIdx.x < 5) {
    float s = 0.f;
#pragma unroll
    for (int w = 0; w < NWAVES; ++w) s += sm[w][threadIdx.x];
    ws[threadIdx.x * NBLOCKS + blockIdx.x] = s;
  }
}

// ---------------------------------------------------------------------------
// Kernel 2: finalize. 5 full waves, one per loss component. Each wave folds
// its 1024 partials through chained V_WMMA_F32_16X16X4_F32 with B = ones:
// D[m][n] = sum_k A[m][k] + C[m][n] -> 64 f32 values reduced per WMMA in
// full f32 precision. Lane mapping into A is irrelevant (everything sums),
// so A operands load as b64 pairs directly.
// ---------------------------------------------------------------------------
__global__ void yolo_final(const float* __restrict__ ws, float* __restrict__ out,
                           float invN) {
  __shared__ float red[5];
  int lane = threadIdx.x & 31;
  int comp = threadIdx.x >> 5;  // 0..4, uniform per wave -> EXEC all-ones

  const v2f* src = reinterpret_cast<const v2f*>(ws + comp * NBLOCKS);
  v8f acc = {};
  v2f ones;
  ones.x = 1.0f;
  ones.y = 1.0f;

  for (int i = 0; i < NBLOCKS / 2; i += 32) {
    v2f a = src[i + lane];  // 64 consecutive partials per wave-iteration
    // 8 args: (neg_a, A, neg_b, B, c_mod, C, reuse_a, reuse_b)
    acc = __builtin_amdgcn_wmma_f32_16x16x4_f32(false, a, false, ones,
                                                (short)0, acc, false, false);
  }

  // per-lane: rows 0..7 (lanes 0-15) or rows 8..15 (lanes 16-31) of D
  float s = acc[0] + acc[1] + acc[2] + acc[3] + acc[4] + acc[5] + acc[6] + acc[7];
  s += __shfl_xor(s, 16, 32);  // combine the two half-wave row groups
  if (lane == 0) red[comp] = s;
  __syncthreads();

  if (threadIdx.x == 0) {
    float loc = red[0], con = red[1], noo = red[2], cls = red[3], ncf = red[4];
    float total = (5.0f * loc + 2.0f * con + ncf + 0.5f * noo + cls) * invN;
    out[0] = total;
    out[1] = loc;
    out[2] = con;
    out[3] = noo;
    out[4] = cls;
  }
}

// ---------------------------------------------------------------------------
extern "C" void kernel_launch(void* const* d_in, const int* in_sizes, int n_in,
                              void* d_out, int out_size, void* d_ws, size_t ws_size,
                              hipStream_t stream) {
  const float* pred = (const float*)d_in[0];
  const float* targ = (const float*)d_in[1];
  float* ws = (float*)d_ws;  // 5 * NBLOCKS floats = 20 KB
  int ncells = in_sizes[0] / 30;             // B * S * S
  float invN = 1.0f / (float)(ncells / 49);  // N = B

  yolo_partial<<<NBLOCKS, BLOCK, 0, stream>>>(pred, targ, ws, ncells);
  yolo_final<<<1, 160, 0, stream>>>(ws, (float*)d_out, invN);
}